// DecoderBlock_43456479101356
// MI455X (gfx1250) — compile-verified
//
#include <hip/hip_runtime.h>

// ---------------------------------------------------------------------------
// GAT (12 heads, 128->32) + segment softmax + MLP, for MI455X (gfx1250).
// Dense GEMMs use V_WMMA_F32_16X16X4_F32 (wave32 WMMA, fp32 precision).
// ---------------------------------------------------------------------------

typedef __attribute__((ext_vector_type(2))) float v2f;
typedef __attribute__((ext_vector_type(8))) float v8f;

#define HEADS     12
#define IN_C      128
#define OUT_C     32
#define HID       64            // 2*OUT_C
#define WCOLS     384           // HEADS*OUT_C
#define NEG_SLOPE 0.2f
#define ENC_NEG_INF 0x007FFFFFu // enc(-inf)

__device__ __forceinline__ v8f wmma_f32(v2f a, v2f b, v8f c) {
    // (neg_a, A, neg_b, B, c_mod, C, reuse_a, reuse_b)
    return __builtin_amdgcn_wmma_f32_16x16x4_f32(false, a, false, b,
                                                 (short)0, c, false, false);
}

// Order-preserving float->uint encoding for atomicMax-based segment max.
__device__ __forceinline__ unsigned enc_f32(float f) {
    unsigned u = __float_as_uint(f);
    return (u & 0x80000000u) ? ~u : (u | 0x80000000u);
}
__device__ __forceinline__ float dec_f32(unsigned u) {
    return (u & 0x80000000u) ? __uint_as_float(u & 0x7FFFFFFFu)
                             : __uint_as_float(~u);
}

__device__ __forceinline__ void edge_sd(const int* __restrict__ ei, int E,
                                        int e, int& s, int& d) {
    if (e < E) { s = ei[e]; d = ei[E + e]; }
    else       { s = d = e - E; }            // self loops appended
}

// ---------------------------------------------------------------------------
// Init scratch: segment-max = -inf, denom = 0, y accumulator = 0.
// ---------------------------------------------------------------------------
__global__ void k_init(unsigned* __restrict__ m_enc, float* __restrict__ denom,
                       float* __restrict__ yacc, int N) {
    int t = blockIdx.x * blockDim.x + threadIdx.x;
    if (t < N * HEADS) { m_enc[t] = ENC_NEG_INF; denom[t] = 0.0f; }
    if (t < N * OUT_C) { yacc[t] = 0.0f; }
}

// ---------------------------------------------------------------------------
// h[N,384] = x[N,128] @ W[128,384], fp32 WMMA.
// Block = 256 thr (8 waves). Block owns one 16-row M tile (x tile in LDS),
// each wave sweeps 3 of the 24 N-tiles; 32 k-steps of 16x16x4 each.
// ---------------------------------------------------------------------------
__global__ __launch_bounds__(256) void k_gemm_h(const float* __restrict__ x,
                                                const float* __restrict__ W,
                                                float* __restrict__ h, int N) {
    __shared__ float xs[16 * IN_C];                // 8 KB
    const int mbase = blockIdx.x * 16;
    const int tid   = threadIdx.x;

    for (int i = tid; i < 16 * IN_C; i += 256) {
        int r = i / IN_C, c = i % IN_C;
        int row = mbase + r;
        xs[i] = (row < N) ? x[row * IN_C + c] : 0.0f;
    }
    __syncthreads();

    const int wave = tid >> 5;
    const int lane = tid & 31;
    const int m    = lane & 15;      // A-matrix row (lanes 0-15 / 16-31)
    const int kp   = lane >> 4;      // K pair select (ISA 16x4 A layout)

    for (int nt = wave; nt < WCOLS / 16; nt += 8) {
        const int col = nt * 16 + (lane & 15);
        v8f acc = {};
        for (int kk = 0; kk < IN_C / 4; ++kk) {
            const int k0 = kk * 4 + 2 * kp;
            v2f a; a.x = xs[m * IN_C + k0];
                   a.y = xs[m * IN_C + k0 + 1];
            v2f b; b.x = W[k0 * WCOLS + col];
                   b.y = W[(k0 + 1) * WCOLS + col];
            acc = wmma_f32(a, b, acc);
        }
        #pragma unroll
        for (int v = 0; v < 8; ++v) {              // C/D: VGPR v -> M = v (+8)
            const int row = mbase + v + 8 * kp;
            if (row < N) h[row * WCOLS + col] = acc[v];
        }
    }
}

// ---------------------------------------------------------------------------
// Attention coefficients: a_src/a_dst[n,h] = <h[n,h,:], att_*[h,:]>.
// ---------------------------------------------------------------------------
__global__ void k_att(const float* __restrict__ h,
                      const float* __restrict__ att_src,
                      const float* __restrict__ att_dst,
                      float* __restrict__ a_src, float* __restrict__ a_dst,
                      int N) {
    int t = blockIdx.x * blockDim.x + threadIdx.x;
    if (t >= N * HEADS) return;
    int head = t % HEADS;
    const float* hp = h + (t / HEADS) * WCOLS + head * OUT_C;
    const float* as = att_src + head * OUT_C;
    const float* ad = att_dst + head * OUT_C;
    float s = 0.0f, d = 0.0f;
    #pragma unroll
    for (int c = 0; c < OUT_C; ++c) { float v = hp[c]; s += v * as[c]; d += v * ad[c]; }
    a_src[t] = s;
    a_dst[t] = d;
}

// ---------------------------------------------------------------------------
// Segment max over incoming edges (leaky-relu'd logits) via atomicMax(uint).
// ---------------------------------------------------------------------------
__global__ void k_edge_max(const int* __restrict__ ei,
                           const float* __restrict__ a_src,
                           const float* __restrict__ a_dst,
                           unsigned* __restrict__ m_enc, int E, int NE) {
    int t = blockIdx.x * blockDim.x + threadIdx.x;
    if (t >= NE * HEADS) return;
    int e = t / HEADS, hd = t % HEADS;
    int s, d; edge_sd(ei, E, e, s, d);
    float ev = a_src[s * HEADS + hd] + a_dst[d * HEADS + hd];
    ev = (ev > 0.0f) ? ev : NEG_SLOPE * ev;
    atomicMax(&m_enc[d * HEADS + hd], enc_f32(ev));
}

// ---------------------------------------------------------------------------
// Segment sum of exp(e - max) per (dst, head).
// ---------------------------------------------------------------------------
__global__ void k_edge_den(const int* __restrict__ ei,
                           const float* __restrict__ a_src,
                           const float* __restrict__ a_dst,
                           const unsigned* __restrict__ m_enc,
                           float* __restrict__ denom, int E, int NE) {
    int t = blockIdx.x * blockDim.x + threadIdx.x;
    if (t >= NE * HEADS) return;
    int e = t / HEADS, hd = t % HEADS;
    int s, d; edge_sd(ei, E, e, s, d);
    int idx = d * HEADS + hd;
    float ev = a_src[s * HEADS + hd] + a_dst[idx];
    ev = (ev > 0.0f) ? ev : NEG_SLOPE * ev;
    atomicAdd(&denom[idx], __expf(ev - dec_f32(m_enc[idx])));
}

// ---------------------------------------------------------------------------
// Aggregation, head-mean folded in (mean is linear):
//   yacc[dst,c] += (1/12) * sum_h alpha[e,h] * h[src,h,c]
// One wave32 per edge; lane = channel c. Lanes 0..11 compute alpha_h,
// broadcast via __shfl; 12-head reduction in registers -> ONE atomic/(e,c).
// h is L2-resident (76.8 MB < 192 MB) so the gather runs at L2 bandwidth.
// ---------------------------------------------------------------------------
__global__ __launch_bounds__(256) void k_edge_agg(const int* __restrict__ ei,
                                                  const float* __restrict__ a_src,
                                                  const float* __restrict__ a_dst,
                                                  const unsigned* __restrict__ m_enc,
                                                  const float* __restrict__ denom,
                                                  const float* __restrict__ h,
                                                  float* __restrict__ yacc,
                                                  int E, int NE) {
    int t    = blockIdx.x * blockDim.x + threadIdx.x;
    int e    = t >> 5;                 // uniform per wave
    int lane = t & 31;
    if (e >= NE) return;               // whole wave exits together

    int s, d; edge_sd(ei, E, e, s, d);

    float alpha = 0.0f;
    if (lane < HEADS) {
        int idx = d * HEADS + lane;
        float ev = a_src[s * HEADS + lane] + a_dst[idx];
        ev = (ev > 0.0f) ? ev : NEG_SLOPE * ev;
        float ex = __expf(ev - dec_f32(m_enc[idx]));
        alpha = ex / (denom[idx] + 1e-16f);
    }

    const float* hp = h + s * WCOLS;
    float acc = 0.0f;
    #pragma unroll
    for (int hd = 0; hd < HEADS; ++hd) {
        float ah = __shfl(alpha, hd, 32);
        acc += ah * hp[hd * OUT_C + lane];         // coalesced 128B line
    }
    atomicAdd(&yacc[d * OUT_C + lane], acc * (1.0f / HEADS));
}

// ---------------------------------------------------------------------------
// Finalize: y = relu(yacc + bias); z = relu(y@W1+b1)@W2+b2; out = relu(y+z).
// Block = 256 thr (8 waves); each wave owns 16 nodes, tiles in LDS,
// both MLP GEMMs on fp32 WMMA (K=32 -> 8 steps, K=64 -> 16 steps).
// ---------------------------------------------------------------------------
__global__ __launch_bounds__(256) void k_mlp(const float* __restrict__ yacc,
                                             const float* __restrict__ bias,
                                             const float* __restrict__ W1,
                                             const float* __restrict__ b1,
                                             const float* __restrict__ W2,
                                             const float* __restrict__ b2,
                                             float* __restrict__ out, int N) {
    __shared__ float y_s[8][16][OUT_C];   // 16 KB
    __shared__ float z_s[8][16][HID];     // 32 KB
    const int tid  = threadIdx.x;
    const int w    = tid >> 5;
    const int lane = tid & 31;
    const int base = blockIdx.x * 128 + w * 16;

    // y tile: relu(yacc + bias)
    for (int i = 0; i < 16; ++i) {
        int node = base + i;
        float v = 0.0f;
        if (node < N) {
            v = yacc[node * OUT_C + lane] + bias[lane];
            v = (v > 0.0f) ? v : 0.0f;
        }
        y_s[w][i][lane] = v;
    }
    __syncthreads();

    const int m  = lane & 15;
    const int kp = lane >> 4;
    const int nl = lane & 15;

    // z1 = relu(y @ W1 + b1)   [16 x 64]
    for (int nt = 0; nt < HID / 16; ++nt) {
        const int col = nt * 16 + nl;
        const float bv = b1[col];
        v8f acc = {bv, bv, bv, bv, bv, bv, bv, bv};
        for (int kk = 0; kk < OUT_C / 4; ++kk) {
            const int k0 = kk * 4 + 2 * kp;
            v2f a; a.x = y_s[w][m][k0];
                   a.y = y_s[w][m][k0 + 1];
            v2f b; b.x = W1[k0 * HID + col];
                   b.y = W1[(k0 + 1) * HID + col];
            acc = wmma_f32(a, b, acc);
        }
        #pragma unroll
        for (int v = 0; v < 8; ++v) {
            float z = acc[v];
            z_s[w][v + 8 * kp][col] = (z > 0.0f) ? z : 0.0f;
        }
    }
    __syncthreads();

    // z = z1 @ W2 + b2, out = relu(y + z)   [16 x 32]
    for (int nt = 0; nt < OUT_C / 16; ++nt) {
        const int col = nt * 16 + nl;
        const float bv = b2[col];
        v8f acc = {bv, bv, bv, bv, bv, bv, bv, bv};
        for (int kk = 0; kk < HID / 4; ++kk) {
            const int k0 = kk * 4 + 2 * kp;
            v2f a; a.x = z_s[w][m][k0];
                   a.y = z_s[w][m][k0 + 1];
            v2f b; b.x = W2[k0 * OUT_C + col];
                   b.y = W2[(k0 + 1) * OUT_C + col];
            acc = wmma_f32(a, b, acc);
        }
        #pragma unroll
        for (int v = 0; v < 8; ++v) {
            const int row  = v + 8 * kp;
            const int node = base + row;
            if (node < N) {
                float r = y_s[w][row][col] + acc[v];
                out[node * OUT_C + col] = (r > 0.0f) ? r : 0.0f;
            }
        }
    }
}

// ---------------------------------------------------------------------------
extern "C" void kernel_launch(void* const* d_in, const int* in_sizes, int n_in,
                              void* d_out, int out_size, void* d_ws, size_t ws_size,
                              hipStream_t stream) {
    const float* x       = (const float*)d_in[0];
    const int*   ei      = (const int*)  d_in[1];   // edge_index [2,E]
    const float* W       = (const float*)d_in[2];
    const float* att_src = (const float*)d_in[3];
    const float* att_dst = (const float*)d_in[4];
    const float* bias    = (const float*)d_in[5];
    const float* W1      = (const float*)d_in[6];
    const float* b1      = (const float*)d_in[7];
    const float* W2      = (const float*)d_in[8];
    const float* b2      = (const float*)d_in[9];
    float* out = (float*)d_out;

    const int N  = in_sizes[0] / IN_C;   // 50000
    const int E  = in_sizes[1] / 2;      // 800000
    const int NE = E + N;                // edges + self loops

    // Workspace carve-up (~93 MB total)
    char* ws = (char*)d_ws;
    float*    h_buf = (float*)ws;    ws += (size_t)N * WCOLS * sizeof(float);
    float*    a_src = (float*)ws;    ws += (size_t)N * HEADS * sizeof(float);
    float*    a_dst = (float*)ws;    ws += (size_t)N * HEADS * sizeof(float);
    unsigned* m_enc = (unsigned*)ws; ws += (size_t)N * HEADS * sizeof(unsigned);
    float*    denom = (float*)ws;    ws += (size_t)N * HEADS * sizeof(float);
    float*    yacc  = (float*)ws;    ws += (size_t)N * OUT_C * sizeof(float);

    k_init<<<(N * OUT_C + 255) / 256, 256, 0, stream>>>(m_enc, denom, yacc, N);
    k_gemm_h<<<(N + 15) / 16, 256, 0, stream>>>(x, W, h_buf, N);
    k_att<<<(N * HEADS + 255) / 256, 256, 0, stream>>>(h_buf, att_src, att_dst,
                                                       a_src, a_dst, N);
    k_edge_max<<<(NE * HEADS + 255) / 256, 256, 0, stream>>>(ei, a_src, a_dst,
                                                             m_enc, E, NE);
    k_edge_den<<<(NE * HEADS + 255) / 256, 256, 0, stream>>>(ei, a_src, a_dst,
                                                             m_enc, denom, E, NE);
    k_edge_agg<<<((size_t)NE * 32 + 255) / 256, 256, 0, stream>>>(ei, a_src, a_dst,
                                                                  m_enc, denom,
                                                                  h_buf, yacc, E, NE);
    k_mlp<<<(N + 127) / 128, 256, 0, stream>>>(yacc, bias, W1, b1, W2, b2, out, N);
}